// DenseGINEConv_12352325943905
// MI455X (gfx1250) — compile-verified
//
#include <hip/hip_runtime.h>

#define B_ 4
#define N_ 512
#define F_ 64
#define H_ 128
#define NEG 0.01f

typedef float v2f __attribute__((ext_vector_type(2)));
typedef float v8f __attribute__((ext_vector_type(8)));

__device__ __forceinline__ float lrelu(float t) { return fmaxf(t, NEG * t); }

// One block per (b, 16-row j-tile). 512 threads = 16 wave32.
// Phase 1: stream edge_attr once (float4, coalesced), accumulate
//          agg[j,f] = sum_i leaky(adj[i,j]*(x[i,f]+e[i,j,f])) split over 2 i-halves.
// Phase 2: out = x + agg  (16x64 tile in LDS, stride 68 => conflict-free A frags)
// Phase 3: h1 = leaky(out@W1 + b1)  via fp32 WMMA 16x16x4, waves 0..7 (one 16-col tile each)
// Phase 4: y  = h1@W2 + b2, masked  via fp32 WMMA, waves 8..11
__global__ __launch_bounds__(512) void gine_fused_kernel(
    const float* __restrict__ x, const float* __restrict__ adj,
    const float* __restrict__ ea, const int* __restrict__ mask,
    const float* __restrict__ W1, const float* __restrict__ b1,
    const float* __restrict__ W2, const float* __restrict__ b2,
    float* __restrict__ out)
{
    __shared__ float s_adj[2][16 * 16];   // [i-chunk row][j]
    __shared__ float s_x[2][16 * 64];     // [i-chunk row][f]
    __shared__ float s_out[16 * 68];      // 16x64 out tile, padded stride 68
    __shared__ float s_h1[16 * 132];      // 16x128 hidden tile, padded stride 132

    const int tid = threadIdx.x;
    const int g   = tid >> 8;     // i-half group (0 or 1)
    const int t   = tid & 255;
    const int jj  = t >> 4;       // staging: i-offset; compute: j-offset in tile
    const int f4  = t & 15;       // staging: f-quad / j-offset; compute: f-quad
    const int b   = blockIdx.y;
    const int j0  = blockIdx.x * 16;

    float acc0 = 0.f, acc1 = 0.f, acc2 = 0.f, acc3 = 0.f;

    const int iBase = g * (N_ / 2);
    for (int ch = 0; ch < (N_ / 2) / 16; ++ch) {
        const int ic = iBase + ch * 16;
        // Stage adj (16 i x 16 j) and x (16 i x 64 f) for this chunk.
        s_adj[g][jj * 16 + f4] = adj[(b * N_ + (ic + jj)) * N_ + j0 + f4];
        {
            const float4 xv = *(const float4*)&x[(b * N_ + (ic + jj)) * F_ + f4 * 4];
            *(float4*)&s_x[g][jj * 64 + f4 * 4] = xv;
        }
        __syncthreads();
#pragma unroll
        for (int ii = 0; ii < 16; ++ii) {
            const int i = ic + ii;
            const float4 e4 = *(const float4*)&ea[((b * N_ + i) * N_ + (j0 + jj)) * F_ + f4 * 4];
            const float  a  = s_adj[g][ii * 16 + jj];
            const float4 xv = *(const float4*)&s_x[g][ii * 64 + f4 * 4];
            float u;
            u = a * (xv.x + e4.x); acc0 += lrelu(u);
            u = a * (xv.y + e4.y); acc1 += lrelu(u);
            u = a * (xv.z + e4.z); acc2 += lrelu(u);
            u = a * (xv.w + e4.w); acc3 += lrelu(u);
        }
        __syncthreads();
    }

    // Combine the two i-half partials and add the self-loop x[b, j0+jj, :].
    if (g == 1) {
        s_out[jj * 68 + f4 * 4 + 0] = acc0;
        s_out[jj * 68 + f4 * 4 + 1] = acc1;
        s_out[jj * 68 + f4 * 4 + 2] = acc2;
        s_out[jj * 68 + f4 * 4 + 3] = acc3;
    }
    __syncthreads();
    if (g == 0) {
        const float4 xs = *(const float4*)&x[(b * N_ + (j0 + jj)) * F_ + f4 * 4];
        s_out[jj * 68 + f4 * 4 + 0] += acc0 + xs.x;
        s_out[jj * 68 + f4 * 4 + 1] += acc1 + xs.y;
        s_out[jj * 68 + f4 * 4 + 2] += acc2 + xs.z;
        s_out[jj * 68 + f4 * 4 + 3] += acc3 + xs.w;
    }
    __syncthreads();

    const int wave = tid >> 5;
    const int lane = tid & 31;
    const int half = lane >> 4;
    const int lrow = lane & 15;

    // GEMM1: (16x64) @ (64x128) -> 16x128, fp32 WMMA 16x16x4, waves 0..7.
    if (wave < 8) {
        const int col = wave * 16 + lrow;          // column in H
        v8f c = {0.f, 0.f, 0.f, 0.f, 0.f, 0.f, 0.f, 0.f};
#pragma unroll
        for (int kk = 0; kk < F_ / 4; ++kk) {
            const int kb = kk * 4 + half * 2;
            v2f a2, bf;
            a2.x = s_out[lrow * 68 + kb];
            a2.y = s_out[lrow * 68 + kb + 1];
            bf.x = W1[kb * H_ + col];
            bf.y = W1[(kb + 1) * H_ + col];
            c = __builtin_amdgcn_wmma_f32_16x16x4_f32(
                    false, a2, false, bf, (short)0, c, false, false);
        }
        const float bias = b1[col];
#pragma unroll
        for (int v = 0; v < 8; ++v) {
            s_h1[(v + half * 8) * 132 + col] = lrelu(c[v] + bias);
        }
    }
    __syncthreads();

    // GEMM2: (16x128) @ (128x64) -> 16x64, fp32 WMMA, waves 8..11; mask + store.
    if (wave >= 8 && wave < 12) {
        const int col = (wave - 8) * 16 + lrow;    // column in F
        v8f c = {0.f, 0.f, 0.f, 0.f, 0.f, 0.f, 0.f, 0.f};
#pragma unroll
        for (int kk = 0; kk < H_ / 4; ++kk) {
            const int kb = kk * 4 + half * 2;
            v2f a2, bf;
            a2.x = s_h1[lrow * 132 + kb];
            a2.y = s_h1[lrow * 132 + kb + 1];
            bf.x = W2[kb * F_ + col];
            bf.y = W2[(kb + 1) * F_ + col];
            c = __builtin_amdgcn_wmma_f32_16x16x4_f32(
                    false, a2, false, bf, (short)0, c, false, false);
        }
        const float bias = b2[col];
#pragma unroll
        for (int v = 0; v < 8; ++v) {
            const int row = v + half * 8;
            const int j   = j0 + row;
            const int m   = mask[b * N_ + j];
            out[(b * N_ + j) * F_ + col] = m ? (c[v] + bias) : 0.0f;
        }
    }
}

extern "C" void kernel_launch(void* const* d_in, const int* in_sizes, int n_in,
                              void* d_out, int out_size, void* d_ws, size_t ws_size,
                              hipStream_t stream) {
    (void)in_sizes; (void)n_in; (void)d_ws; (void)ws_size; (void)out_size;
    const float* x    = (const float*)d_in[0];
    const float* adj  = (const float*)d_in[1];
    const float* ea   = (const float*)d_in[2];
    const int*   mask = (const int*)d_in[3];
    const float* W1   = (const float*)d_in[4];
    const float* b1   = (const float*)d_in[5];
    const float* W2   = (const float*)d_in[6];
    const float* b2   = (const float*)d_in[7];

    dim3 grid(N_ / 16, B_);
    gine_fused_kernel<<<grid, 512, 0, stream>>>(x, adj, ea, mask, W1, b1, W2, b2,
                                                (float*)d_out);
}